// PairPairSupConLoss_21019569947027
// MI455X (gfx1250) — compile-verified
//
#include <hip/hip_runtime.h>
#include <math.h>

// PairPairSupConLoss on MI455X (gfx1250).
//
// Algebraic reduction: with f row-L2-normalized,
//   raw_self[i,j]  = (M1 + G) / (sqrt(M2 + 2*M3 + 1) * T)
//   raw_other[i,j] = (M3 + G) / (sqrt(M4 + 2*M5 + 1) * T)
// where G = f f^T, M1 = f^4 f^T, M2 = f^6 (f^2)^T, M3 = f^3 (f^2)^T,
//       M4 = f^4 (f^4)^T, M5 = f^2 (f^3)^T  -- six FP32 GEMMs (512x512x256)
// executed with V_WMMA_F32_16X16X4_F32 (the FP32 tensor op; f16 paths would
// underflow the f^6 terms). Then
//   vs_all[i] = sum_j (i!=j) * (exp(raw_self) + 3*exp(raw_other))
//   loss = -(1/B^2) * sum_i ( sum_j raw_self[i,j] - B*log(vs_all[i]) )

typedef __attribute__((ext_vector_type(2))) float v2f;
typedef __attribute__((ext_vector_type(8))) float v8f;

#define BB 512
#define DD 256
#define TEMP 0.07f
#define TILES (BB / 16)   // 32 tiles per dimension

// ---------------- Kernel 1: row L2 normalization ----------------
__global__ void k_normalize(const float* __restrict__ x, float* __restrict__ fn) {
    __shared__ float red[DD];
    const int row = blockIdx.x;
    const int t = threadIdx.x;
    float v = x[row * DD + t];
    red[t] = v * v;
    __syncthreads();
    for (int s = DD / 2; s > 0; s >>= 1) {
        if (t < s) red[t] += red[t + s];
        __syncthreads();
    }
    const float inv = 1.0f / fmaxf(sqrtf(red[0]), 1e-12f);
    fn[row * DD + t] = v * inv;
}

// ---------------- Kernel 2: six fused FP32 WMMA GEMMs + logits ----------------
// One wave32 per 16x16 output tile. A-fragment (16x4 f32): lane L holds row
// M=L%16; VGPR0/1 hold K = 2*(L/16) + {0,1}. B-fragment mirrors (col-major
// access of f^T => rows of f for tile j). C/D: VGPR r -> M = r + 8*(L/16),
// N = L%16.
__global__ void k_pair_tiles(const float* __restrict__ fn,
                             float* __restrict__ pE,
                             float* __restrict__ pR) {
    const int tile = blockIdx.x;
    const int ti = tile / TILES;
    const int tj = tile % TILES;
    const int lane = threadIdx.x;       // 0..31, full wave active (EXEC all 1s)
    const int mn   = lane & 15;
    const int half = lane >> 4;

    const float* __restrict__ arow = fn + (ti * 16 + mn) * DD;
    const float* __restrict__ brow = fn + (tj * 16 + mn) * DD;

    v8f accG  = {}; v8f accM1 = {}; v8f accM2 = {};
    v8f accM3 = {}; v8f accM4 = {}; v8f accM5 = {};

#pragma unroll 4
    for (int k0 = 0; k0 < DD; k0 += 4) {
        const int ka = k0 + 2 * half;
        const float2 av = *(const float2*)(arow + ka);
        const float2 bv = *(const float2*)(brow + ka);
        const float a0 = av.x, a1 = av.y, b0 = bv.x, b1 = bv.y;
        // powers in registers (cheap VALU; avoids 4 extra streams from memory)
        const float a0_2 = a0 * a0,   a1_2 = a1 * a1;
        const float b0_2 = b0 * b0,   b1_2 = b1 * b1;
        const float a0_3 = a0_2 * a0, a1_3 = a1_2 * a1;
        const float b0_3 = b0_2 * b0, b1_3 = b1_2 * b1;
        const float a0_4 = a0_2 * a0_2, a1_4 = a1_2 * a1_2;
        const float b0_4 = b0_2 * b0_2, b1_4 = b1_2 * b1_2;
        const float a0_6 = a0_3 * a0_3, a1_6 = a1_3 * a1_3;

        const v2f A1 = {a0,   a1};   const v2f B1 = {b0,   b1};
        const v2f A2 = {a0_2, a1_2}; const v2f B2 = {b0_2, b1_2};
        const v2f A3 = {a0_3, a1_3}; const v2f B3 = {b0_3, b1_3};
        const v2f A4 = {a0_4, a1_4}; const v2f B4 = {b0_4, b1_4};
        const v2f A6 = {a0_6, a1_6};

        accG  = __builtin_amdgcn_wmma_f32_16x16x4_f32(false, A1, false, B1, (short)0, accG,  false, false);
        accM1 = __builtin_amdgcn_wmma_f32_16x16x4_f32(false, A4, false, B1, (short)0, accM1, false, false);
        accM2 = __builtin_amdgcn_wmma_f32_16x16x4_f32(false, A6, false, B2, (short)0, accM2, false, false);
        accM3 = __builtin_amdgcn_wmma_f32_16x16x4_f32(false, A3, false, B2, (short)0, accM3, false, false);
        accM4 = __builtin_amdgcn_wmma_f32_16x16x4_f32(false, A4, false, B4, (short)0, accM4, false, false);
        accM5 = __builtin_amdgcn_wmma_f32_16x16x4_f32(false, A2, false, B3, (short)0, accM5, false, false);
    }

    // Epilogue: logits, exp terms, per-row partial sums over this j-tile.
#pragma unroll
    for (int r = 0; r < 8; ++r) {
        const int Mi = r + 8 * half;
        const int i = ti * 16 + Mi;
        const int j = tj * 16 + mn;
        const float G  = accG[r];
        const float m1 = accM1[r], m2 = accM2[r], m3 = accM3[r];
        const float m4 = accM4[r], m5 = accM5[r];
        const float n1 = fmaxf(sqrtf(m2 + 2.0f * m3 + 1.0f), 1e-12f);
        const float n2 = fmaxf(sqrtf(m4 + 2.0f * m5 + 1.0f), 1e-12f);
        const float raw_self  = (m1 + G) / (n1 * TEMP);
        const float raw_other = (m3 + G) / (n2 * TEMP);
        const float e = (i != j) ? (expf(raw_self) + 3.0f * expf(raw_other)) : 0.0f;
        // sum over the 16 columns of this tile: xor-shuffle within each half
        float sE = e, sR = raw_self;
        for (int m = 1; m < 16; m <<= 1) {
            sE += __shfl_xor(sE, m, 32);
            sR += __shfl_xor(sR, m, 32);
        }
        if (mn == 0) {   // lane 0 (M=r) and lane 16 (M=r+8) hold the sums
            pE[tj * BB + i] = sE;
            pR[tj * BB + i] = sR;
        }
    }
}

// ---------------- Kernel 3: deterministic final reduction ----------------
__global__ void k_finalize(const float* __restrict__ pE,
                           const float* __restrict__ pR,
                           float* __restrict__ out) {
    __shared__ float red[BB];
    const int i = threadIdx.x;   // 0..511 => one row each
    float sE = 0.0f, sR = 0.0f;
    for (int t = 0; t < TILES; ++t) {   // fixed order -> deterministic
        sE += pE[t * BB + i];
        sR += pR[t * BB + i];
    }
    red[i] = sR - (float)BB * logf(sE);
    __syncthreads();
    for (int s = BB / 2; s > 0; s >>= 1) {
        if (i < s) red[i] += red[i + s];
        __syncthreads();
    }
    if (i == 0) out[0] = -red[0] / ((float)BB * (float)BB);
}

extern "C" void kernel_launch(void* const* d_in, const int* in_sizes, int n_in,
                              void* d_out, int out_size, void* d_ws, size_t ws_size,
                              hipStream_t stream) {
    (void)in_sizes; (void)n_in; (void)out_size; (void)ws_size;
    const float* feats = (const float*)d_in[0];
    float* out = (float*)d_out;

    float* fn = (float*)d_ws;            // 512*256 f32 normalized features
    float* pE = fn + BB * DD;            // 32*512 f32 partial exp-sums
    float* pR = pE + TILES * BB;         // 32*512 f32 partial raw_self sums

    k_normalize  <<<BB, DD, 0, stream>>>(feats, fn);
    k_pair_tiles <<<TILES * TILES, 32, 0, stream>>>(fn, pE, pR);
    k_finalize   <<<1, BB, 0, stream>>>(pE, pR, out);
}